// Block_ltpe_68272800137801
// MI455X (gfx1250) — compile-verified
//
#include <hip/hip_runtime.h>
#include <stdint.h>

#define IMG_W 1024
#define IMG_H 1024
#define ROWS  8            // output rows per block
#define TPB   256          // 8 wave32 per block

typedef __attribute__((address_space(3))) float lds_f;
typedef float __attribute__((ext_vector_type(4))) fvec4;   // native vector for NT store

// ---------------------------------------------------------------------------
// CDNA5 async global->LDS DMA (ASYNCcnt domain). Each lane copies 16 bytes
// from its global address to its LDS address. gfx1250-only instruction.
// ---------------------------------------------------------------------------
__device__ __forceinline__ void async_copy_b128(float* lds_dst, const float* g_src) {
    lds_f* lp = (lds_f*)lds_dst;                       // addrspacecast: generic -> LDS
    unsigned lds_off = (unsigned)(uintptr_t)lp;        // byte offset within WG LDS alloc
    asm volatile("global_load_async_to_lds_b128 %0, %1, off"
                 :: "v"(lds_off), "v"(g_src)
                 : "memory");
}

__device__ __forceinline__ void wait_async_all() {
    asm volatile("s_wait_asynccnt 0" ::: "memory");
}

// Stage rows [row0-1, row0+ROWS] of one image into LDS (zero-fill out-of-range
// rows to honor the reference's zero padding). tile is (ROWS+2) x 1024 floats.
__device__ __forceinline__ void load_tile(float* tile, const float* img,
                                          int row0, int tid) {
#pragma unroll
    for (int i = 0; i < ROWS + 2; ++i) {
        int gy = row0 - 1 + i;                        // uniform across the block
        float* dst = &tile[i * IMG_W + tid * 4];
        if (gy >= 0 && gy < IMG_H) {
            async_copy_b128(dst, img + (size_t)gy * IMG_W + tid * 4);
        } else {
            dst[0] = 0.f; dst[1] = 0.f; dst[2] = 0.f; dst[3] = 0.f;
        }
    }
    wait_async_all();
    __syncthreads();
}

// Fused 3x3 stencil: out_pre = 0.5 + 0.5*(c - sum_j w_j * neighbor_j),
// with zero contributions outside the image (left/right handled here,
// top/bottom handled by the zero-filled halo rows in LDS).
__device__ __forceinline__ float stencil_at(const float* t, int tr, int x) {
    const float W0 = 1.f/255.f,  W1 = 2.f/255.f,  W2 = 4.f/255.f,  W3 = 8.f/255.f;
    const float W4 = 16.f/255.f, W5 = 32.f/255.f, W6 = 64.f/255.f, W7 = 128.f/255.f;
    const float* rU = t + (tr - 1) * IMG_W;   // dy = -1
    const float* rC = t + tr * IMG_W;         // dy =  0
    const float* rD = t + (tr + 1) * IMG_W;   // dy = +1
    float c = rC[x];
    float U = rU[x], D = rD[x];
    bool hasL = (x > 0), hasR = (x < IMG_W - 1);
    float L  = hasL ? rC[x - 1] : 0.f;
    float UL = hasL ? rU[x - 1] : 0.f;
    float DL = hasL ? rD[x - 1] : 0.f;
    float R  = hasR ? rC[x + 1] : 0.f;
    float UR = hasR ? rU[x + 1] : 0.f;
    float DR = hasR ? rD[x + 1] : 0.f;
    float acc = W0 * L;              // j0 (0,-1)
    acc = fmaf(W1, DL, acc);         // j1 (1,-1)
    acc = fmaf(W2, D,  acc);         // j2 (1, 0)
    acc = fmaf(W3, DR, acc);         // j3 (1, 1)
    acc = fmaf(W4, R,  acc);         // j4 (0, 1)
    acc = fmaf(W5, UR, acc);         // j5 (-1,1)
    acc = fmaf(W6, U,  acc);         // j6 (-1,0)
    acc = fmaf(W7, UL, acc);         // j7 (-1,-1)
    return fmaf(0.5f, c - acc, 0.5f);
}

// ---------------------------------------------------------------------------
// Pass 0: zero the f64 accumulators in workspace.
// ---------------------------------------------------------------------------
__global__ void ws_zero_kernel(double* acc, int n) {
    int i = threadIdx.x + blockIdx.x * blockDim.x;
    if (i < n) acc[i] = 0.0;
}

// ---------------------------------------------------------------------------
// Pass 1: stencil + per-image sum / sum-of-squares (reads x once; x lines
// are installed RT in L2 so the apply pass re-reads hit the 192MB L2).
// ---------------------------------------------------------------------------
__global__ void __launch_bounds__(TPB)
ltpe_reduce_kernel(const float* __restrict__ x, double* __restrict__ acc, int B) {
    __shared__ float tile[(ROWS + 2) * IMG_W];
    __shared__ float red[16];
    int tid  = threadIdx.x;
    int b    = blockIdx.y;
    int row0 = blockIdx.x * ROWS;
    const float* img = x + (size_t)b * ((size_t)IMG_H * IMG_W);

    load_tile(tile, img, row0, tid);

    float s = 0.f, s2 = 0.f;
#pragma unroll
    for (int i = 0; i < (ROWS * IMG_W) / (TPB * 4); ++i) {   // 8 float4 per thread
        int q = i * TPB + tid;          // float4 index within tile
        int r = q >> 8;                 // 256 float4 per row
        int xc = (q & 255) * 4;
#pragma unroll
        for (int k = 0; k < 4; ++k) {
            float v = stencil_at(tile, r + 1, xc + k);
            s += v;
            s2 = fmaf(v, v, s2);
        }
    }
    // wave32 tree reduction
#pragma unroll
    for (int off = 16; off > 0; off >>= 1) {
        s  += __shfl_down(s,  off, 32);
        s2 += __shfl_down(s2, off, 32);
    }
    int wv = tid >> 5, ln = tid & 31;
    if (ln == 0) { red[wv] = s; red[8 + wv] = s2; }
    __syncthreads();
    if (tid == 0) {
        double ds = 0.0, ds2 = 0.0;
#pragma unroll
        for (int i = 0; i < TPB / 32; ++i) { ds += (double)red[i]; ds2 += (double)red[8 + i]; }
        atomicAdd(&acc[b],     ds);    // global_atomic_add_f64
        atomicAdd(&acc[B + b], ds2);
    }
}

// ---------------------------------------------------------------------------
// Pass 2: tiny finalize — mean and rsqrt(var+eps) per image in f64.
// ---------------------------------------------------------------------------
__global__ void ltpe_finalize_kernel(const double* __restrict__ acc,
                                     float* __restrict__ ms, int B) {
    int b = threadIdx.x + blockIdx.x * blockDim.x;
    if (b < B) {
        const double N = (double)IMG_H * (double)IMG_W;
        double mean = acc[b] / N;
        double var  = acc[B + b] / N - mean * mean;   // biased variance
        double scale = rsqrt(var + 1e-5);
        ms[2 * b]     = (float)mean;
        ms[2 * b + 1] = (float)scale;
    }
}

// ---------------------------------------------------------------------------
// Pass 3: recompute stencil (x re-read hits L2), normalize, and store the
// output NON-TEMPORALLY so the 128MB write stream does not evict x from L2.
// ---------------------------------------------------------------------------
__global__ void __launch_bounds__(TPB)
ltpe_apply_kernel(const float* __restrict__ x, const float* __restrict__ ms,
                  float* __restrict__ out) {
    __shared__ float tile[(ROWS + 2) * IMG_W];
    int tid  = threadIdx.x;
    int b    = blockIdx.y;
    int row0 = blockIdx.x * ROWS;
    const float* img = x + (size_t)b * ((size_t)IMG_H * IMG_W);
    float mean  = ms[2 * b];
    float scale = ms[2 * b + 1];

    load_tile(tile, img, row0, tid);

    float* oimg = out + (size_t)b * ((size_t)IMG_H * IMG_W);
#pragma unroll
    for (int i = 0; i < (ROWS * IMG_W) / (TPB * 4); ++i) {
        int q = i * TPB + tid;
        int r = q >> 8;
        int xc = (q & 255) * 4;
        fvec4 o;
        o.x = (stencil_at(tile, r + 1, xc + 0) - mean) * scale;
        o.y = (stencil_at(tile, r + 1, xc + 1) - mean) * scale;
        o.z = (stencil_at(tile, r + 1, xc + 2) - mean) * scale;
        o.w = (stencil_at(tile, r + 1, xc + 3) - mean) * scale;
        // global_store_b128 ... th:TH_STORE_NT — streaming store, keep x in L2
        __builtin_nontemporal_store(o,
            reinterpret_cast<fvec4*>(&oimg[(size_t)(row0 + r) * IMG_W + xc]));
    }
}

// ---------------------------------------------------------------------------
extern "C" void kernel_launch(void* const* d_in, const int* in_sizes, int n_in,
                              void* d_out, int out_size, void* d_ws, size_t ws_size,
                              hipStream_t stream) {
    const float* x = (const float*)d_in[0];
    float* out = (float*)d_out;
    int B = in_sizes[0] / (IMG_H * IMG_W);            // 32

    double* acc = (double*)d_ws;                      // [0..B): sum, [B..2B): sumsq
    float*  ms  = (float*)((char*)d_ws + (size_t)2 * B * sizeof(double));

    ws_zero_kernel<<<1, 64, 0, stream>>>(acc, 2 * B);

    dim3 grid(IMG_H / ROWS, B);                       // 128 x 32 blocks
    ltpe_reduce_kernel<<<grid, TPB, 0, stream>>>(x, acc, B);
    ltpe_finalize_kernel<<<1, B, 0, stream>>>(acc, ms, B);
    ltpe_apply_kernel<<<grid, TPB, 0, stream>>>(x, ms, out);
}